// DMGI_69947837383507
// MI455X (gfx1250) — compile-verified
//
#include <hip/hip_runtime.h>

#define F_IN  128
#define D_OUT 64
#define R_REL 3
#define LXS   130   // LDS row stride for x tile: odd*2 -> conflict-free, 8B-aligned for even k

typedef float v2f __attribute__((ext_vector_type(2)));
typedef float v8f __attribute__((ext_vector_type(8)));

// ---------------------------------------------------------------------------
// GEMM: xw[N,64] = x[N,128] @ W[128,64], fp32 WMMA (V_WMMA_F32_16X16X4_F32).
// Block = 128 threads (4 waves). Block computes a 64-row x 64-col tile.
// Wave w computes rows [w*16, w*16+16) across all 4 N-tiles of 16 columns.
// x tile staged in LDS (coalesced global loads, padded stride for ds reads);
// W is 32KB and stays L2-resident, read directly per fragment.
// ---------------------------------------------------------------------------
__global__ __launch_bounds__(128)
void gemm_xw_kernel(const float* __restrict__ x, const float* __restrict__ W,
                    float* __restrict__ xw, int n)
{
    __shared__ float lx[64 * LXS];
    const int tid = threadIdx.x;
    const int r0  = blockIdx.x * 64;

    // Cooperative load of 64x128 x-tile (rows beyond N zero-filled).
    for (int idx = tid; idx < 64 * F_IN; idx += 128) {
        int rr = idx >> 7;       // local row
        int kk = idx & 127;      // k
        int grow = r0 + rr;
        lx[rr * LXS + kk] = (grow < n) ? x[(size_t)grow * F_IN + kk] : 0.0f;
    }
    __syncthreads();

    const int wave = tid >> 5;
    const int lane = tid & 31;
    const int half = lane >> 4;   // which 16-lane half
    const int mr   = lane & 15;   // row-in-tile (A) / col-in-tile (B, C)
    const int m0   = wave * 16;   // wave's row tile inside block

    v8f c[4] = {v8f{}, v8f{}, v8f{}, v8f{}};

    #pragma unroll
    for (int k0 = 0; k0 < F_IN; k0 += 4) {
        const int ka = k0 + 2 * half;          // this half-wave's K pair base
        // A fragment: lane holds x[m0+mr][ka], x[m0+mr][ka+1]
        const float* ap = &lx[(m0 + mr) * LXS + ka];
        v2f a; a.x = ap[0]; a.y = ap[1];
        #pragma unroll
        for (int t = 0; t < 4; ++t) {
            // B fragment: lane holds W[ka][t*16+mr], W[ka+1][t*16+mr]
            const float* bp = &W[(size_t)ka * D_OUT + t * 16 + mr];
            v2f b; b.x = bp[0]; b.y = bp[D_OUT];
            c[t] = __builtin_amdgcn_wmma_f32_16x16x4_f32(
                false, a, false, b, (short)0, c[t], false, false);
        }
    }

    // C layout: VGPR v -> M = v + 8*half, N = mr (+ 16*t)
    #pragma unroll
    for (int t = 0; t < 4; ++t) {
        #pragma unroll
        for (int v = 0; v < 8; ++v) {
            int grow = r0 + m0 + v + 8 * half;
            if (grow < n)
                xw[(size_t)grow * D_OUT + t * 16 + mr] = c[t][v];
        }
    }
}

// ---------------------------------------------------------------------------
// Utility: fill a float buffer with a constant.
// ---------------------------------------------------------------------------
__global__ void fill_kernel(float* __restrict__ p, float v, int n)
{
    int i = blockIdx.x * 256 + threadIdx.x;
    if (i < n) p[i] = v;
}

// deg[col[e]] += 1  (float counts are exact for these magnitudes)
__global__ void deg_kernel(const int* __restrict__ col, float* __restrict__ deg, int e)
{
    int i = blockIdx.x * 256 + threadIdx.x;
    if (i < e) atomicAdd(&deg[col[i]], 1.0f);
}

// dis = deg^-0.5 (deg >= 1 always due to self-loops)
__global__ void rsqrt_kernel(float* __restrict__ d, int n)
{
    int i = blockIdx.x * 256 + threadIdx.x;
    if (i < n) d[i] = rsqrtf(d[i]);
}

// Self-loop term initializes both output views:
//   pos[i,:] = xw[i,:]        * dis[i]^2
//   neg[i,:] = xw[nidx[i],:]  * dis[i]^2   (x[neg_idx] @ W == xw[neg_idx])
__global__ __launch_bounds__(256)
void selfloop_kernel(const float* __restrict__ xw, const float* __restrict__ dis,
                     const int* __restrict__ nidx,
                     float* __restrict__ pos, float* __restrict__ neg, int n)
{
    int i = blockIdx.x * 256 + threadIdx.x;
    if (i >= n * D_OUT) return;
    int node = i >> 6;
    int d    = i & 63;
    float s  = dis[node]; s = s * s;
    pos[i] = xw[i] * s;
    neg[i] = xw[(size_t)nidx[node] * D_OUT + d] * s;
}

// Edge scatter, one wave per edge, lane = feature pair (2 floats -> 64 feats).
// Gathers 256B/wave contiguous from xw, atomic-adds 256B/wave into out rows.
__global__ __launch_bounds__(256)
void scatter_kernel(const float* __restrict__ xw, const float* __restrict__ dis,
                    const int* __restrict__ row, const int* __restrict__ col,
                    const int* __restrict__ nidx,
                    float* __restrict__ pos, float* __restrict__ neg, int e)
{
    int eidx = blockIdx.x * 8 + (threadIdx.x >> 5);
    int lane = threadIdx.x & 31;
    if (eidx >= e) return;

    int rs = row[eidx];
    int cd = col[eidx];
    float norm = dis[rs] * dis[cd];
    int f = lane * 2;

    float2 vp = *(const float2*)&xw[(size_t)rs * D_OUT + f];
    float* pd = &pos[(size_t)cd * D_OUT + f];
    atomicAdd(pd + 0, vp.x * norm);
    atomicAdd(pd + 1, vp.y * norm);

    int rn = nidx[rs];
    float2 vn = *(const float2*)&xw[(size_t)rn * D_OUT + f];
    float* nd = &neg[(size_t)cd * D_OUT + f];
    atomicAdd(nd + 0, vn.x * norm);
    atomicAdd(nd + 1, vn.y * norm);
}

// Finalize: h = relu(out + b) in place for both views; accumulate pos-view
// column mean via LDS ds_add_f32 reduction + one global atomic per feature.
__global__ __launch_bounds__(256)
void finalize_kernel(const float* __restrict__ bias,
                     float* __restrict__ pos, float* __restrict__ neg,
                     float* __restrict__ summ, int n, float invn)
{
    __shared__ float s[D_OUT];
    int t = threadIdx.x;
    if (t < D_OUT) s[t] = 0.0f;
    __syncthreads();

    int node = blockIdx.x * 4 + (t >> 6);
    int d    = t & 63;
    if (node < n) {
        size_t idx = (size_t)node * D_OUT + d;
        float bv = bias[d];
        float pv = pos[idx] + bv; pv = pv > 0.0f ? pv : 0.0f; pos[idx] = pv;
        float nv = neg[idx] + bv; nv = nv > 0.0f ? nv : 0.0f; neg[idx] = nv;
        atomicAdd(&s[d], pv);            // LDS atomic (ds_add_f32)
    }
    __syncthreads();
    if (t < D_OUT) atomicAdd(&summ[t], s[t] * invn);
}

// ---------------------------------------------------------------------------
// Launch. Workspace layout: xw[N*64] floats | dis[N] floats  (~26 MB, reused
// across the 3 relations via stream ordering).
// d_out layout: pos[R,N,64] | neg[R,N,64] | summaries[R,64].
// ---------------------------------------------------------------------------
extern "C" void kernel_launch(void* const* d_in, const int* in_sizes, int n_in,
                              void* d_out, int out_size, void* d_ws, size_t ws_size,
                              hipStream_t stream)
{
    (void)n_in; (void)out_size; (void)ws_size;
    const float* x  = (const float*)d_in[0];
    const int*   ei = (const int*)  d_in[1];
    const int*   ni = (const int*)  d_in[2];
    const float* W  = (const float*)d_in[3];
    const float* b  = (const float*)d_in[4];

    const int n = in_sizes[0] / F_IN;
    const int e = in_sizes[1] / (R_REL * 2);

    float* out  = (float*)d_out;
    float* xw   = (float*)d_ws;
    float* dis  = xw + (size_t)n * D_OUT;
    const size_t nd = (size_t)n * D_OUT;
    float* summ_base = out + 2 * R_REL * nd;

    // Zero all summary slots once per call (harness poisons d_out).
    fill_kernel<<<(R_REL * D_OUT + 255) / 256, 256, 0, stream>>>(summ_base, 0.0f, R_REL * D_OUT);

    for (int r = 0; r < R_REL; ++r) {
        const int*   rowp = ei + (size_t)r * 2 * e;
        const int*   colp = rowp + e;
        const int*   nidx = ni + (size_t)r * n;
        const float* Wr   = W + (size_t)r * F_IN * D_OUT;
        const float* br   = b + (size_t)r * D_OUT;
        float* pos  = out + (size_t)r * nd;
        float* neg  = out + (size_t)(R_REL + r) * nd;
        float* summ = summ_base + (size_t)r * D_OUT;

        gemm_xw_kernel<<<(n + 63) / 64, 128, 0, stream>>>(x, Wr, xw, n);
        fill_kernel  <<<(n + 255) / 256, 256, 0, stream>>>(dis, 1.0f, n);   // self-loop degree
        deg_kernel   <<<(e + 255) / 256, 256, 0, stream>>>(colp, dis, e);
        rsqrt_kernel <<<(n + 255) / 256, 256, 0, stream>>>(dis, n);
        selfloop_kernel<<<((size_t)n * D_OUT + 255) / 256, 256, 0, stream>>>(xw, dis, nidx, pos, neg, n);
        scatter_kernel <<<(e + 7) / 8, 256, 0, stream>>>(xw, dis, rowp, colp, nidx, pos, neg, e);
        finalize_kernel<<<(n + 3) / 4, 256, 0, stream>>>(br, pos, neg, summ, n, 1.0f / (float)n);
    }
}